// RandomAddressModule_59356448031032
// MI455X (gfx1250) — compile-verified
//
#include <hip/hip_runtime.h>
#include <stdint.h>
#include <stddef.h>

// ---------------------------------------------------------------------------
// RandomHash one-hot scatter for MI455X (gfx1250).
//
// out[d, i, s] = (s == ((a_d*x_i + b_d) mod (2^31-1)) mod S) ? 1.0f : 0.0f
// D=8, B=4096, S=16384  ->  2 GiB fp32 output: pure HBM-store-bound (~92us
// floor at 23.3 TB/s). No matrix math -> WMMA inapplicable; the win is wide
// NT b128 stores, full coalescing, block-uniform scalar hash math, and a
// divergence-free single-owner fixup store for the 1.0.
// ---------------------------------------------------------------------------

#define MERSENNE_P 2147483647u // 2^31 - 1

struct HashParams {
  unsigned long long a[8];
  unsigned long long b[8];
};

// ============================ host-side RNG ================================
// Bit-exact replication of:
//   rng = np.random.default_rng(1)
//   a = rng.integers(1, P, size=8)   # Lemire bounded sampling, 32-bit path
//   b = rng.integers(0, P, size=8)
namespace nprng {

struct PCG64 {
  unsigned __int128 state;
  unsigned __int128 inc;
  bool has32;
  uint32_t buf32;

  static unsigned __int128 mult() {
    return (((unsigned __int128)0x2360ed051fc65da4ULL) << 64) |
           0x4385df649fccf645ULL;
  }
  void step() { state = state * mult() + inc; }
  uint64_t out() const {
    uint64_t hi = (uint64_t)(state >> 64);
    uint64_t lo = (uint64_t)state;
    uint64_t v = hi ^ lo;
    unsigned rot = (unsigned)(hi >> 58);
    return (v >> rot) | (v << ((64u - rot) & 63u)); // rotr64 (rot=0 safe)
  }
  uint64_t next64() { step(); return out(); }
  uint32_t next32() { // numpy buffers 64-bit draws into two 32-bit outputs
    if (has32) { has32 = false; return buf32; }
    uint64_t v = next64();
    has32 = true;
    buf32 = (uint32_t)(v >> 32);
    return (uint32_t)v;
  }
  void seed_from(const uint64_t s[4]) { // pcg64_srandom_r
    unsigned __int128 initstate = (((unsigned __int128)s[0]) << 64) | s[1];
    unsigned __int128 initseq   = (((unsigned __int128)s[2]) << 64) | s[3];
    state = 0;
    inc = (initseq << 1) | 1;
    step();
    state += initstate;
    step();
    has32 = false;
    buf32 = 0;
  }
};

// SeedSequence(1).generate_state(4, uint64)
static void seedseq1_state4(uint64_t out[4]) {
  const uint32_t INIT_A = 0x43b0d7e5u, MULT_A = 0x931e8875u;
  const uint32_t INIT_B = 0x8b51f9ddu, MULT_B = 0x58f38dedu;
  const uint32_t MIX_L  = 0xca01f9ddu, MIX_R  = 0x4973f715u;

  uint32_t hc = INIT_A;
  auto hashmix = [&hc, MULT_A](uint32_t v) {
    v ^= hc; hc *= MULT_A; v *= hc; v ^= v >> 16; return v;
  };
  auto mix = [MIX_L, MIX_R](uint32_t x, uint32_t y) {
    uint32_t r = x * MIX_L - y * MIX_R; r ^= r >> 16; return r;
  };

  const uint32_t entropy[1] = {1u}; // seed = 1, empty spawn key
  uint32_t pool[4];
  for (int i = 0; i < 4; ++i) pool[i] = hashmix(i < 1 ? entropy[i] : 0u);
  for (int s = 0; s < 4; ++s)
    for (int d = 0; d < 4; ++d)
      if (s != d) pool[d] = mix(pool[d], hashmix(pool[s]));
  // no leftover entropy (len(entropy)=1 <= pool size 4)

  uint32_t hb = INIT_B;
  uint32_t w[8];
  for (int i = 0; i < 8; ++i) {
    uint32_t dv = pool[i & 3];
    dv ^= hb; hb *= MULT_B; dv *= hb; dv ^= dv >> 16;
    w[i] = dv;
  }
  for (int i = 0; i < 4; ++i)
    out[i] = (uint64_t)w[2 * i] | ((uint64_t)w[2 * i + 1] << 32);
}

// numpy bounded_lemire_uint32: rng_inclusive = high - low - 1 (< 2^32 path)
static uint32_t lemire_u32(PCG64& g, uint32_t rng_inclusive) {
  const uint32_t rng_excl = rng_inclusive + 1u;
  uint64_t m = (uint64_t)g.next32() * (uint64_t)rng_excl;
  uint32_t leftover = (uint32_t)m;
  if (leftover < rng_excl) {
    const uint32_t threshold = (uint32_t)((0xFFFFFFFFu - rng_inclusive) % rng_excl);
    while (leftover < threshold) {
      m = (uint64_t)g.next32() * (uint64_t)rng_excl;
      leftover = (uint32_t)m;
    }
  }
  return (uint32_t)(m >> 32);
}

static HashParams make_hash_params() {
  uint64_t s[4];
  seedseq1_state4(s);
  PCG64 g;
  g.seed_from(s);
  HashParams hp;
  for (int d = 0; d < 8; ++d) hp.a[d] = 1ull + lemire_u32(g, MERSENNE_P - 2u); // integers(1,P)
  for (int d = 0; d < 8; ++d) hp.b[d] = (unsigned long long)lemire_u32(g, MERSENNE_P - 1u); // integers(0,P)
  return hp;
}

} // namespace nprng

// ============================== device side ================================

typedef float v4f __attribute__((ext_vector_type(4)));

__device__ __forceinline__ unsigned hash_mod_p(unsigned long long a,
                                               unsigned long long b,
                                               unsigned long long xv) {
  unsigned long long m = a * xv + b; // a < 2^31, x < 10^6 -> m < 2^52
  // mod (2^31 - 1) via Mersenne folding: 2^31 == 1 (mod P)
  m = (m & 0x7FFFFFFFull) + (m >> 31);
  m = (m & 0x7FFFFFFFull) + (m >> 31);
  if (m >= 0x7FFFFFFFull) m -= 0x7FFFFFFFull;
  return (unsigned)m;
}

// Specialized path: SLOT known at compile time (16384 from setup_inputs()).
// 2D grid: blockIdx.x = i (batch), blockIdx.y = d (dep) -> no integer division.
// Each 256-thread block (8 wave32s) owns one contiguous 64 KB row:
//   - 16 fully-unrolled back-to-back global_store_b128 (TH=NT) of a constant
//     zero vector: consecutive lanes -> consecutive 16 B -> 512 B/wave bursts,
//     16 outstanding stores per thread to saturate the HBM write path.
//   - the single thread owning idx's 16 B chunk then stores 1.0f; per-wave
//     stores are kept in order with stores, so it lands after its own zero.
template <int SLOT>
__global__ __launch_bounds__(256) void onehot_fixed_kernel(
    const long long* __restrict__ x, float* __restrict__ out, HashParams hp) {
  const unsigned i = blockIdx.x;
  const unsigned d = blockIdx.y;
  const unsigned batch = gridDim.x;

  const unsigned long long xv = (unsigned long long)x[i]; // block-uniform
  const unsigned idx = hash_mod_p(hp.a[d], hp.b[d], xv) % (unsigned)SLOT;

  float* __restrict__ rowp = out + ((size_t)d * batch + i) * (size_t)SLOT;
  const unsigned tid = threadIdx.x;

  const v4f zero = {0.0f, 0.0f, 0.0f, 0.0f};
  constexpr int NV = SLOT / 4; // float4 chunks per row
#pragma unroll
  for (int k = 0; k < NV / 256; ++k) {
    __builtin_nontemporal_store(
        zero, reinterpret_cast<v4f*>(rowp) + (k * 256 + tid));
  }
  // chunk v = k*256 + tid  ->  owner tid of chunk (idx>>2) is (idx>>2) & 255
  if (tid == ((idx >> 2) & 255u)) {
    __builtin_nontemporal_store(1.0f, rowp + idx);
  }
}

// Generic fallback for unexpected shapes (runtime slot_num, branch-free select).
__global__ __launch_bounds__(256) void onehot_generic_kernel(
    const long long* __restrict__ x, float* __restrict__ out, HashParams hp,
    int slot_num) {
  const unsigned i = blockIdx.x;
  const unsigned d = blockIdx.y;
  const unsigned batch = gridDim.x;

  const unsigned long long xv = (unsigned long long)x[i];
  const unsigned idx = hash_mod_p(hp.a[d], hp.b[d], xv) % (unsigned)slot_num;

  float* __restrict__ rowp = out + ((size_t)d * batch + i) * (size_t)slot_num;
  const int nvec = slot_num >> 2; // requires slot_num % 4 == 0

#pragma unroll 4
  for (int v = threadIdx.x; v < nvec; v += 256) {
    const unsigned c0 = (unsigned)v << 2;
    v4f val;
    val.x = (c0 + 0u == idx) ? 1.0f : 0.0f;
    val.y = (c0 + 1u == idx) ? 1.0f : 0.0f;
    val.z = (c0 + 2u == idx) ? 1.0f : 0.0f;
    val.w = (c0 + 3u == idx) ? 1.0f : 0.0f;
    __builtin_nontemporal_store(val, reinterpret_cast<v4f*>(rowp) + v);
  }
}

// ================================ launch ===================================

extern "C" void kernel_launch(void* const* d_in, const int* in_sizes, int n_in,
                              void* d_out, int out_size, void* d_ws, size_t ws_size,
                              hipStream_t stream) {
  (void)n_in; (void)d_ws; (void)ws_size;

  const long long* x = (const long long*)d_in[0]; // input_tensor, int64
  const int batch = in_sizes[0];                  // 4096

  // dep_dim / slot_num are device-resident scalars; setup_inputs() fixes
  // dep_dim = 8. slot_num is derived from the output footprint.
  const int dep_dim = 8;
  const long long rows_ll = (long long)dep_dim * (long long)batch;
  const int slot_num = (rows_ll > 0) ? (int)((long long)out_size / rows_ll) : 0; // 16384
  if (slot_num <= 0 || batch <= 0) return;

  const HashParams hp = nprng::make_hash_params(); // deterministic every call

  const dim3 grid((unsigned)batch, (unsigned)dep_dim); // 4096 x 8 blocks
  if (slot_num == 16384) {
    onehot_fixed_kernel<16384><<<grid, 256, 0, stream>>>(x, (float*)d_out, hp);
  } else {
    onehot_generic_kernel<<<grid, 256, 0, stream>>>(x, (float*)d_out, hp,
                                                    slot_num);
  }
}